// PointNet2FPModule_19550691131525
// MI455X (gfx1250) — compile-verified
//
#include <hip/hip_runtime.h>

// ---------------------------------------------------------------------------
// PointNet++ FP module for MI455X (gfx1250, wave32).
// Memory-bound (~200MB @ 23.3TB/s => ~9us floor); GEMMs run on the fp32 WMMA
// path (v_wmma_f32_16x16x4_f32) keeping full fp32 precision at ~ the same
// cost as the bandwidth floor.
// ---------------------------------------------------------------------------

typedef __attribute__((ext_vector_type(2))) float v2f;
typedef __attribute__((ext_vector_type(8))) float v8f;

#define B_    2
#define N_    16384      // unknown points per batch (2^14)
#define M_    4096       // known points per batch
#define C1_   128        // skip channels (2^7)
#define C2_   256        // known-feature channels
#define CIN1  (C1_ + C2_)   // 384
#define CMID  256
#define COUT  256
#define NCOLS (B_ * N_)     // 32768 GEMM columns

// ---------------------------------------------------------------------------
// Kernel 1: 3-NN + inverse-distance weights. Known points cached in LDS.
// ---------------------------------------------------------------------------
__global__ __launch_bounds__(256)
void knn_kernel(const float* __restrict__ unknown, const float* __restrict__ known,
                int* __restrict__ idx, float* __restrict__ wgt)
{
    __shared__ float kp[M_ * 3];     // 48KB of 320KB LDS
    const int b = blockIdx.y;
    const float* kb = known + (size_t)b * M_ * 3;
    for (int t = threadIdx.x; t < M_ * 3; t += 256) kp[t] = kb[t];
    __syncthreads();

    const int i  = blockIdx.x * 256 + threadIdx.x;
    const int gi = b * N_ + i;
    const float qx = unknown[gi * 3 + 0];
    const float qy = unknown[gi * 3 + 1];
    const float qz = unknown[gi * 3 + 2];

    float d0 = 1e30f, d1 = 1e30f, d2 = 1e30f;
    int   i0 = 0,     i1 = 0,     i2 = 0;
    for (int j = 0; j < M_; ++j) {
        const float dx = kp[3 * j + 0] - qx;
        const float dy = kp[3 * j + 1] - qy;
        const float dz = kp[3 * j + 2] - qz;
        const float d = dx * dx + dy * dy + dz * dz;
        if (d < d2) {
            if (d < d1) {
                if (d < d0) { d2 = d1; i2 = i1; d1 = d0; i1 = i0; d0 = d; i0 = j; }
                else        { d2 = d1; i2 = i1; d1 = d;  i1 = j; }
            } else          { d2 = d;  i2 = j; }
        }
    }
    const float r0 = 1.0f / (sqrtf(fmaxf(d0, 0.0f)) + 1e-8f);
    const float r1 = 1.0f / (sqrtf(fmaxf(d1, 0.0f)) + 1e-8f);
    const float r2 = 1.0f / (sqrtf(fmaxf(d2, 0.0f)) + 1e-8f);
    const float rs = 1.0f / (r0 + r1 + r2);
    idx[gi * 3 + 0] = i0; idx[gi * 3 + 1] = i1; idx[gi * 3 + 2] = i2;
    wgt[gi * 3 + 0] = r0 * rs; wgt[gi * 3 + 1] = r1 * rs; wgt[gi * 3 + 2] = r2 * rs;
}

// ---------------------------------------------------------------------------
// Kernel 2: transpose known_feats (B,C2,m) -> (B,m,C2) for coalesced gathers.
// ---------------------------------------------------------------------------
__global__ __launch_bounds__(256)
void transpose_kf_kernel(const float* __restrict__ in, float* __restrict__ out)
{
    __shared__ float tile[32][33];
    const int b  = blockIdx.z;
    const int c0 = blockIdx.y * 32;   // C2/32 = 8
    const int j0 = blockIdx.x * 32;   // m/32  = 128
    for (int r = threadIdx.y; r < 32; r += 8)
        tile[r][threadIdx.x] = in[(size_t)b * C2_ * M_ + (size_t)(c0 + r) * M_ + j0 + threadIdx.x];
    __syncthreads();
    for (int r = threadIdx.y; r < 32; r += 8)
        out[(size_t)b * M_ * C2_ + (size_t)(j0 + r) * C2_ + c0 + threadIdx.x] = tile[threadIdx.x][r];
}

// ---------------------------------------------------------------------------
// Kernel 3: gather + weighted blend -> F1 rows [0,256) (channel-major),
// with LDS transpose so the global writes are contiguous per channel row.
// Block handles 32 points x 256 channels.
// ---------------------------------------------------------------------------
__global__ __launch_bounds__(256)
void interp_kernel(const int* __restrict__ idx, const float* __restrict__ wgt,
                   const float* __restrict__ kfT, float* __restrict__ F1)
{
    __shared__ float tile[256][33];
    const int p0 = blockIdx.x * 32;
    const int c  = threadIdx.x;
    const int b  = p0 >> 14;                       // 32 | 16384 => one batch per block
    const float* base = kfT + (size_t)b * M_ * C2_;
    for (int pp = 0; pp < 32; ++pp) {
        const int p = p0 + pp;
        const int j0 = idx[p * 3 + 0], j1 = idx[p * 3 + 1], j2 = idx[p * 3 + 2];
        const float w0 = wgt[p * 3 + 0], w1 = wgt[p * 3 + 1], w2 = wgt[p * 3 + 2];
        tile[c][pp] = w0 * base[(size_t)j0 * C2_ + c]
                    + w1 * base[(size_t)j1 * C2_ + c]
                    + w2 * base[(size_t)j2 * C2_ + c];
    }
    __syncthreads();
#pragma unroll
    for (int r = 0; r < 32; ++r) {
        const int linear = threadIdx.x + r * 256;
        const int cc = linear >> 5, pp = linear & 31;
        F1[(size_t)cc * NCOLS + p0 + pp] = tile[cc][pp];
    }
}

// ---------------------------------------------------------------------------
// Kernel 4: copy skip features into F1 rows [256,384). Fully coalesced.
// ---------------------------------------------------------------------------
__global__ __launch_bounds__(256)
void copy_skip_kernel(const float* __restrict__ uf, float* __restrict__ F1)
{
    const int t = blockIdx.x * 256 + threadIdx.x;  // 0 .. B*C1*N-1
    const int i = t & (N_ - 1);
    const int c = (t >> 14) & (C1_ - 1);
    const int b = t >> 21;
    F1[(size_t)(C2_ + c) * NCOLS + b * N_ + i] = uf[t];
}

// ---------------------------------------------------------------------------
// Kernel 5: fold BatchNorm(eval) into per-channel scale/bias.
// ---------------------------------------------------------------------------
__global__ void bn_prep_kernel(const float* __restrict__ g, const float* __restrict__ be,
                               const float* __restrict__ mn, const float* __restrict__ vr,
                               float* __restrict__ sc, float* __restrict__ bi)
{
    const int i = threadIdx.x;          // C == 256 == blockDim
    const float s = g[i] * rsqrtf(vr[i] + 1e-5f);
    sc[i] = s;
    bi[i] = be[i] - mn[i] * s;
}

// ---------------------------------------------------------------------------
// Kernel 6: fp32 WMMA GEMM + fused BN + ReLU.
// Out(256 x NCOLS) = A(256 x KDIM) * Bm(KDIM x NCOLS).
// Block tile 64(M) x 128(N), 8 waves as 2x4, each wave 32x32 via 2x2 WMMA
// tiles, K slabs of 32 staged through LDS; K step 4 per v_wmma_f32_16x16x4_f32.
// ---------------------------------------------------------------------------
template <int KDIM, bool FINAL>
__global__ __launch_bounds__(256)
void gemm_bn_relu_kernel(const float* __restrict__ A, const float* __restrict__ Bm,
                         const float* __restrict__ scale, const float* __restrict__ bias,
                         float* __restrict__ Out)
{
    __shared__ float As[64][33];     // padded: odd stride -> conflict-free col reads
    __shared__ float Bs[32][128];

    const int tid    = threadIdx.x;
    const int lane   = tid & 31;
    const int wid    = tid >> 5;
    const int wave_m = wid & 1;       // 0..1
    const int wave_n = wid >> 1;      // 0..3
    const int m0     = blockIdx.y * 64;
    const int n0     = blockIdx.x * 128;
    const int lm     = lane & 15;
    const int khalf  = (lane >> 4) << 1;   // lanes 0-15 -> K+0/1, lanes 16-31 -> K+2/3

    v8f acc[2][2] = {};

    for (int kk = 0; kk < KDIM; kk += 32) {
        __syncthreads();
#pragma unroll
        for (int r = 0; r < 8; ++r) {          // A: 64x32 floats
            const int linear = tid + r * 256;
            const int row = linear >> 5, col = linear & 31;
            As[row][col] = A[(size_t)(m0 + row) * KDIM + kk + col];
        }
#pragma unroll
        for (int r = 0; r < 4; ++r) {          // B: 32x128 floats as float4
            const int linear = tid + r * 256;
            const int row = linear >> 5, c4 = (linear & 31) << 2;
            *(float4*)&Bs[row][c4] =
                *(const float4*)&Bm[(size_t)(kk + row) * NCOLS + n0 + c4];
        }
        __syncthreads();

#pragma unroll
        for (int k4 = 0; k4 < 8; ++k4) {
            const int ks = (k4 << 2) + khalf;
            v2f a0, a1, b0, b1;
            const int am = wave_m * 32 + lm;
            a0.x = As[am][ks];          a0.y = As[am][ks + 1];
            a1.x = As[am + 16][ks];     a1.y = As[am + 16][ks + 1];
            const int bn = wave_n * 32 + lm;
            b0.x = Bs[ks][bn];          b0.y = Bs[ks + 1][bn];
            b1.x = Bs[ks][bn + 16];     b1.y = Bs[ks + 1][bn + 16];
            acc[0][0] = __builtin_amdgcn_wmma_f32_16x16x4_f32(false, a0, false, b0, (short)0, acc[0][0], false, false);
            acc[0][1] = __builtin_amdgcn_wmma_f32_16x16x4_f32(false, a0, false, b1, (short)0, acc[0][1], false, false);
            acc[1][0] = __builtin_amdgcn_wmma_f32_16x16x4_f32(false, a1, false, b0, (short)0, acc[1][0], false, false);
            acc[1][1] = __builtin_amdgcn_wmma_f32_16x16x4_f32(false, a1, false, b1, (short)0, acc[1][1], false, false);
        }
    }

    // Epilogue: D layout = VGPR e, lanes 0-15 -> M=e, lanes 16-31 -> M=8+e.
    const int hi = (lane >> 4) << 3;
#pragma unroll
    for (int tm = 0; tm < 2; ++tm) {
#pragma unroll
        for (int tn = 0; tn < 2; ++tn) {
            const int colg = n0 + wave_n * 32 + tn * 16 + lm;
#pragma unroll
            for (int e = 0; e < 8; ++e) {
                const int o = m0 + wave_m * 32 + tm * 16 + hi + e;
                float v = acc[tm][tn][e] * scale[o] + bias[o];
                v = fmaxf(v, 0.0f);
                if (FINAL) {
                    const int bb = colg >> 14;
                    const int ii = colg & (N_ - 1);
                    Out[(size_t)bb * COUT * N_ + (size_t)o * N_ + ii] = v;
                } else {
                    Out[(size_t)o * NCOLS + colg] = v;
                }
            }
        }
    }
}

// ---------------------------------------------------------------------------
// Host launcher
// ---------------------------------------------------------------------------
extern "C" void kernel_launch(void* const* d_in, const int* in_sizes, int n_in,
                              void* d_out, int out_size, void* d_ws, size_t ws_size,
                              hipStream_t stream)
{
    const float* unknown     = (const float*)d_in[0];
    const float* known       = (const float*)d_in[1];
    const float* unknow_fts  = (const float*)d_in[2];
    const float* known_feats = (const float*)d_in[3];
    const float* W1          = (const float*)d_in[4];
    const float* gamma1      = (const float*)d_in[5];
    const float* beta1       = (const float*)d_in[6];
    const float* mean1       = (const float*)d_in[7];
    const float* var1        = (const float*)d_in[8];
    const float* W2          = (const float*)d_in[9];
    const float* gamma2      = (const float*)d_in[10];
    const float* beta2       = (const float*)d_in[11];
    const float* mean2       = (const float*)d_in[12];
    const float* var2        = (const float*)d_in[13];

    char* ws = (char*)d_ws;
    // workspace layout (bytes, all sizes 256-aligned)
    const size_t OFF_IDX = 0;                              // 2*16384*3*4   = 393216
    const size_t OFF_WGT = OFF_IDX + 393216;               //              = 393216
    const size_t OFF_KFT = OFF_WGT + 393216;               // 2*4096*256*4 = 8388608
    const size_t OFF_F1  = OFF_KFT + 8388608;              // 384*32768*4  = 50331648
    const size_t OFF_H1  = OFF_F1  + 50331648;             // 256*32768*4  = 33554432
    const size_t OFF_S1  = OFF_H1  + 33554432;
    const size_t OFF_B1  = OFF_S1  + 1024;
    const size_t OFF_S2  = OFF_B1  + 1024;
    const size_t OFF_B2  = OFF_S2  + 1024;

    int*   idx    = (int*)  (ws + OFF_IDX);
    float* wgt    = (float*)(ws + OFF_WGT);
    float* kfT    = (float*)(ws + OFF_KFT);
    float* F1     = (float*)(ws + OFF_F1);
    float* H1     = (float*)(ws + OFF_H1);
    float* scale1 = (float*)(ws + OFF_S1);
    float* bias1  = (float*)(ws + OFF_B1);
    float* scale2 = (float*)(ws + OFF_S2);
    float* bias2  = (float*)(ws + OFF_B2);
    float* out    = (float*)d_out;

    // 1) 3-NN + weights
    knn_kernel<<<dim3(N_ / 256, B_), 256, 0, stream>>>(unknown, known, idx, wgt);
    // 2) transpose known_feats for coalesced gathers
    transpose_kf_kernel<<<dim3(M_ / 32, C2_ / 32, B_), dim3(32, 8), 0, stream>>>(known_feats, kfT);
    // 3) interpolate into F1 rows [0,256)
    interp_kernel<<<dim3(NCOLS / 32), 256, 0, stream>>>(idx, wgt, kfT, F1);
    // 4) skip features into F1 rows [256,384)
    copy_skip_kernel<<<dim3((B_ * C1_ * N_) / 256), 256, 0, stream>>>(unknow_fts, F1);
    // 5) fold BN
    bn_prep_kernel<<<1, 256, 0, stream>>>(gamma1, beta1, mean1, var1, scale1, bias1);
    bn_prep_kernel<<<1, 256, 0, stream>>>(gamma2, beta2, mean2, var2, scale2, bias2);
    // 6) WMMA GEMM + BN + ReLU  x2
    gemm_bn_relu_kernel<CIN1, false><<<dim3(NCOLS / 128, CMID / 64), 256, 0, stream>>>(W1, F1, scale1, bias1, H1);
    gemm_bn_relu_kernel<CMID, true ><<<dim3(NCOLS / 128, COUT / 64), 256, 0, stream>>>(W2, H1, scale2, bias2, out);
}